// OverlapTripletLoss1_11991548690924
// MI455X (gfx1250) — compile-verified
//
#include <hip/hip_runtime.h>
#include <math.h>

typedef __attribute__((ext_vector_type(2))) float v2f;
typedef __attribute__((ext_vector_type(8))) float v8f;

#define N_SAMP 65536
#define DDIM   512
#define NCLS   64
#define KOV    64

// ---- workspace layout (in floats) ----
#define CENTERS_OFF 0            // 64*512
#define COUNTS_OFF  32768        // 64 ints
#define C2_OFF      32832        // 64
#define PC_OFF      32896        // 64
#define X2_OFF      32960        // 65536
#define DIST_OFF    98560        // 64*65536 (16 MB)

// ------------------------------------------------------------------
__global__ void zero_kernel(float* p, int n) {
    int i = blockIdx.x * blockDim.x + threadIdx.x;
    if (i < n) p[i] = 0.0f;
}

// per-class sample counts, LDS histogram then global atomics
__global__ void counts_kernel(const int* __restrict__ y, int* __restrict__ counts) {
    __shared__ int hist[NCLS];
    int tid = threadIdx.x;
    if (tid < NCLS) hist[tid] = 0;
    __syncthreads();
    int base = blockIdx.x * 1024;
    for (int i = tid; i < 1024; i += 256)
        atomicAdd(&hist[y[base + i]], 1);
    __syncthreads();
    if (tid < NCLS) atomicAdd(&counts[tid], hist[tid]);
}

// segmented sum of x rows into per-class accumulators.
// grid = (64 row-chunks of 1024 rows, 4 col-chunks of 128 cols), block = 256
__global__ void centers_acc_kernel(const float* __restrict__ x,
                                   const int* __restrict__ y,
                                   float* __restrict__ centers) {
    __shared__ float lacc[NCLS * 128];   // 32 KB
    int tid = threadIdx.x;
    for (int i = tid; i < NCLS * 128; i += 256) lacc[i] = 0.0f;
    __syncthreads();

    int r0    = blockIdx.x * 1024;
    int dbase = blockIdx.y * 128;
    int lane  = tid & 31;
    int rsub  = tid >> 5;                // 8 rows in flight
    for (int it = 0; it < 128; ++it) {
        int row = r0 + it * 8 + rsub;
        int cls = y[row];
        const float4* xp = (const float4*)(x + (size_t)row * DDIM + dbase);
        float4 v = xp[lane];
        float* dst = &lacc[cls * 128 + lane * 4];
        atomicAdd(dst + 0, v.x);
        atomicAdd(dst + 1, v.y);
        atomicAdd(dst + 2, v.z);
        atomicAdd(dst + 3, v.w);
    }
    __syncthreads();
    for (int i = tid; i < NCLS * 128; i += 256) {
        int c = i >> 7, col = i & 127;
        atomicAdd(&centers[c * DDIM + dbase + col], lacc[i]);
    }
}

// centers /= counts ; c2[c] = ||center_c||^2   (one block per class)
__global__ void centers_fin_kernel(float* __restrict__ centers,
                                   const int* __restrict__ counts,
                                   float* __restrict__ c2) {
    __shared__ float red[256];
    int c = blockIdx.x, tid = threadIdx.x;
    float inv = 1.0f / (float)counts[c];
    float acc = 0.0f;
    for (int j = tid; j < DDIM; j += 256) {
        float v = centers[c * DDIM + j] * inv;
        centers[c * DDIM + j] = v;
        acc += v * v;
    }
    red[tid] = acc;
    __syncthreads();
    for (int s = 128; s > 0; s >>= 1) {
        if (tid < s) red[tid] += red[tid + s];
        __syncthreads();
    }
    if (tid == 0) c2[c] = red[0];
}

// x2[n] = ||x_n||^2 ; one wave32 per row, 8 rows per block
__global__ void x2_kernel(const float* __restrict__ x, float* __restrict__ x2) {
    int tid  = threadIdx.x;
    int lane = tid & 31;
    int row  = blockIdx.x * 8 + (tid >> 5);
    const float4* xp = (const float4*)(x + (size_t)row * DDIM);
    float s = 0.0f;
    #pragma unroll
    for (int j = 0; j < 4; ++j) {
        float4 v = xp[lane + j * 32];
        s += v.x * v.x + v.y * v.y + v.z * v.z + v.w * v.w;
    }
    #pragma unroll
    for (int off = 16; off > 0; off >>= 1)
        s += __shfl_xor(s, off, 32);
    if (lane == 0) x2[row] = s;
}

// dist[c][n] = sqrt(max(c2[c] + x2[n] - 2*centers_c.x_n, 1e-12))
// f32 WMMA 16x16x4. Each wave holds ONE B fragment (16 samples of x, the
// streamed 128MB operand -> read exactly once from HBM) and sweeps all four
// 16-class A tiles (centers are 128KB, L2-hot) with 4 accumulators.
// block = 8 waves -> 128 samples x 64 classes; grid = 65536/128 = 512.
__global__ void wmma_dist_kernel(const float* __restrict__ x,
                                 const float* __restrict__ centers,
                                 const float* __restrict__ c2,
                                 const float* __restrict__ x2,
                                 float* __restrict__ dist) {
    int tid   = threadIdx.x;
    int lane  = tid & 31;
    int wave  = tid >> 5;
    int nbase = blockIdx.x * 128 + wave * 16;    // this wave's 16-sample strip
    int half  = lane >> 4;                       // lanes 16-31 hold K+2,K+3
    int l15   = lane & 15;

    // B: 4x16 f32 tile of x^T: B[k][n] = x[n][k] -> contiguous float2 per lane
    const v2f* Bp  = (const v2f*)(x + (size_t)(nbase + l15) * DDIM);
    // A: 16x4 f32 tiles of centers, one pointer per 16-class tile
    const v2f* Ap0 = (const v2f*)(centers + (size_t)(0 * 16 + l15) * DDIM);
    const v2f* Ap1 = (const v2f*)(centers + (size_t)(1 * 16 + l15) * DDIM);
    const v2f* Ap2 = (const v2f*)(centers + (size_t)(2 * 16 + l15) * DDIM);
    const v2f* Ap3 = (const v2f*)(centers + (size_t)(3 * 16 + l15) * DDIM);

    v8f acc0 = {}, acc1 = {}, acc2 = {}, acc3 = {};
    #pragma unroll 2
    for (int kp = 0; kp < DDIM / 2; kp += 2) {   // kp in float2 units, step = K4
        int idx = kp + half;
        v2f b  = Bp[idx];
        v2f a0 = Ap0[idx];
        v2f a1 = Ap1[idx];
        v2f a2 = Ap2[idx];
        v2f a3 = Ap3[idx];
        acc0 = __builtin_amdgcn_wmma_f32_16x16x4_f32(false, a0, false, b, (short)0, acc0, false, false);
        acc1 = __builtin_amdgcn_wmma_f32_16x16x4_f32(false, a1, false, b, (short)0, acc1, false, false);
        acc2 = __builtin_amdgcn_wmma_f32_16x16x4_f32(false, a2, false, b, (short)0, acc2, false, false);
        acc3 = __builtin_amdgcn_wmma_f32_16x16x4_f32(false, a3, false, b, (short)0, acc3, false, false);
    }

    // C/D layout: vgpr i, lanes0-15 -> M=i, N=lane ; lanes16-31 -> M=i+8, N=lane-16
    int ng = nbase + l15;
    float xv = x2[ng];
    v8f accs[4] = {acc0, acc1, acc2, acc3};
    #pragma unroll
    for (int m = 0; m < 4; ++m) {
        #pragma unroll
        for (int i = 0; i < 8; ++i) {
            int cg = m * 16 + i + 8 * half;
            float s  = accs[m][i];
            float d2 = c2[cg] + xv - 2.0f * s;
            dist[(size_t)cg * N_SAMP + ng] = sqrtf(fmaxf(d2, 1e-12f));
        }
    }
}

// exact mean of top-64 (pos: max over members, neg: min over non-members)
// via 4-pass 8-bit radix select on float bits (all dists > 0). One block/class,
// 1024 threads sweeping the L2-resident 256KB dist row.
__global__ void select_kernel(const float* __restrict__ dist,
                              const int* __restrict__ y,
                              float* __restrict__ per_class) {
    __shared__ unsigned cnt[256];
    __shared__ float    bsum[256];
    __shared__ unsigned s_prefix;
    __shared__ int      s_rem;
    __shared__ float    s_sum;
    int c = blockIdx.x, tid = threadIdx.x;
    const float* row = dist + (size_t)c * N_SAMP;
    float means[2];

    for (int mode = 0; mode < 2; ++mode) {       // 0 = pos(max,member), 1 = neg(min,other)
        if (tid == 0) { s_prefix = 0u; s_rem = KOV; s_sum = 0.0f; }
        __syncthreads();
        for (int pass = 0; pass < 4; ++pass) {
            if (tid < 256) { cnt[tid] = 0u; bsum[tid] = 0.0f; }
            __syncthreads();
            unsigned pmask = (pass == 0) ? 0u : (0xFFFFFFFFu << (32 - 8 * pass));
            unsigned pref  = s_prefix;
            int shift = 24 - 8 * pass;
            for (int n = tid; n < N_SAMP; n += 1024) {
                bool mem = (y[n] == c);
                if (mem != (mode == 0)) continue;
                float v = row[n];
                unsigned key = __float_as_uint(v);
                if ((key & pmask) != pref) continue;
                unsigned bin = (key >> shift) & 255u;
                atomicAdd(&cnt[bin], 1u);
                atomicAdd(&bsum[bin], v);
            }
            __syncthreads();
            if (tid == 0) {
                int rem = s_rem; float ss = s_sum; int sel = 0;
                if (mode == 0) {
                    for (int b = 255; b >= 0; --b) {
                        unsigned cb = cnt[b];
                        if (!cb) continue;
                        if ((int)cb < rem) { rem -= (int)cb; ss += bsum[b]; }
                        else { sel = b; break; }
                    }
                } else {
                    for (int b = 0; b < 256; ++b) {
                        unsigned cb = cnt[b];
                        if (!cb) continue;
                        if ((int)cb < rem) { rem -= (int)cb; ss += bsum[b]; }
                        else { sel = b; break; }
                    }
                }
                s_prefix |= (unsigned)sel << shift;
                s_rem = rem; s_sum = ss;
            }
            __syncthreads();
        }
        float T = __uint_as_float(s_prefix);     // exact k-th value (ties fill at T)
        means[mode] = (s_sum + (float)s_rem * T) * (1.0f / (float)KOV);
        __syncthreads();
    }
    if (tid == 0)
        per_class[c] = fmaxf(1.0f + means[0] - means[1], 0.0f);
}

__global__ void final_kernel(const float* __restrict__ per_class, float* __restrict__ out) {
    __shared__ float r[NCLS];
    int tid = threadIdx.x;
    r[tid] = per_class[tid];
    __syncthreads();
    for (int s = 32; s > 0; s >>= 1) {
        if (tid < s) r[tid] += r[tid + s];
        __syncthreads();
    }
    if (tid == 0) out[0] = r[0] / (float)(NCLS * KOV);
}

// ------------------------------------------------------------------
extern "C" void kernel_launch(void* const* d_in, const int* in_sizes, int n_in,
                              void* d_out, int out_size, void* d_ws, size_t ws_size,
                              hipStream_t stream) {
    (void)in_sizes; (void)n_in; (void)out_size; (void)ws_size;
    const float* x = (const float*)d_in[0];
    const int*   y = (const int*)d_in[1];
    float* ws      = (float*)d_ws;
    float* centers = ws + CENTERS_OFF;
    int*   counts  = (int*)(ws + COUNTS_OFF);
    float* c2      = ws + C2_OFF;
    float* pc      = ws + PC_OFF;
    float* x2      = ws + X2_OFF;
    float* dist    = ws + DIST_OFF;
    float* out     = (float*)d_out;

    zero_kernel<<<(32832 + 255) / 256, 256, 0, stream>>>(ws, 32832);
    counts_kernel<<<64, 256, 0, stream>>>(y, counts);
    centers_acc_kernel<<<dim3(64, 4), 256, 0, stream>>>(x, y, centers);
    centers_fin_kernel<<<64, 256, 0, stream>>>(centers, counts, c2);
    x2_kernel<<<N_SAMP / 8, 256, 0, stream>>>(x, x2);
    wmma_dist_kernel<<<N_SAMP / 128, 256, 0, stream>>>(x, centers, c2, x2, dist);
    select_kernel<<<NCLS, 1024, 0, stream>>>(dist, y, pc);
    final_kernel<<<1, NCLS, 0, stream>>>(pc, out);
}